// SemanticOutputLayer_75831942578505
// MI455X (gfx1250) — compile-verified
//
#include <hip/hip_runtime.h>

// ---------------------------------------------------------------------------
// Łukasiewicz clause evaluation over a gathered atom table.
//   out0[i] = min(1, x[a] - x[b] + 1)                      signs (+,-)
//   out1[i] = min(1, x[a] + x[b] - x[c] + 1)               signs (+,+,-)
//   out2[i] = min(1, 2 - x[a] + x[b] + x[c] - x[d])        signs (-,+,+,-)
//
// Memory-bound gather/stream workload (no matrix structure -> no WMMA/TDM):
//   * index streams + output are single-use  -> TH=NT loads/stores
//   * 8 MB gather table is reused 50M times  -> warm it into the 192 MB L2
//     with global_prefetch_b8, gathers use default (RT) policy.
// ---------------------------------------------------------------------------

typedef int   v4i __attribute__((ext_vector_type(4)));
typedef float v2f __attribute__((ext_vector_type(2)));
typedef float v4f __attribute__((ext_vector_type(4)));

static constexpr int BLK = 256;   // 8 wave32 per block

// Warm the gather table into L2 (speculative prefetch; one cacheline/thread).
__global__ void __launch_bounds__(BLK)
prefetch_table_kernel(const float* __restrict__ x, long nbytes) {
  long off = ((long)blockIdx.x * BLK + threadIdx.x) * 128;
  if (off < nbytes) {
    __builtin_prefetch((const char*)x + off, /*rw=*/0, /*locality=*/1);
  }
}

// arity-2, signs (+, -): 1 thread = 2 clauses (one b128 index load).
__global__ void __launch_bounds__(BLK)
clause2_kernel(const float* __restrict__ x, const int* __restrict__ idx,
               float* __restrict__ out, int n) {
  int u  = blockIdx.x * BLK + threadIdx.x;   // unit = 2 clauses
  int c0 = 2 * u;
  if (c0 + 1 < n) {
    v4i id = __builtin_nontemporal_load((const v4i*)idx + u);
    float g0 = x[id.x], g1 = x[id.y], g2 = x[id.z], g3 = x[id.w];
    v2f r;
    r.x = fminf(1.0f, g0 - g1 + 1.0f);
    r.y = fminf(1.0f, g2 - g3 + 1.0f);
    __builtin_nontemporal_store(r, (v2f*)out + u);
  } else if (c0 < n) {                       // scalar tail (n odd)
    int i0 = idx[2 * c0], i1 = idx[2 * c0 + 1];
    out[c0] = fminf(1.0f, x[i0] - x[i1] + 1.0f);
  }
}

// arity-3, signs (+, +, -): 1 thread = 4 clauses (three b128 index loads).
__global__ void __launch_bounds__(BLK)
clause3_kernel(const float* __restrict__ x, const int* __restrict__ idx,
               float* __restrict__ out, int n) {
  int u  = blockIdx.x * BLK + threadIdx.x;   // unit = 4 clauses = 12 ints
  int c0 = 4 * u;
  if (c0 + 3 < n) {
    const v4i* p = (const v4i*)idx + 3 * u;
    v4i a = __builtin_nontemporal_load(p);
    v4i b = __builtin_nontemporal_load(p + 1);
    v4i c = __builtin_nontemporal_load(p + 2);
    v4f r;
    r.x = fminf(1.0f, x[a.x] + x[a.y] - x[a.z] + 1.0f);
    r.y = fminf(1.0f, x[a.w] + x[b.x] - x[b.y] + 1.0f);
    r.z = fminf(1.0f, x[b.z] + x[b.w] - x[c.x] + 1.0f);
    r.w = fminf(1.0f, x[c.y] + x[c.z] - x[c.w] + 1.0f);
    __builtin_nontemporal_store(r, (v4f*)out + u);
  } else {                                   // scalar tail (last unit only)
    for (int cc = c0; cc < n; ++cc) {
      int i0 = idx[3 * cc], i1 = idx[3 * cc + 1], i2 = idx[3 * cc + 2];
      out[cc] = fminf(1.0f, x[i0] + x[i1] - x[i2] + 1.0f);
    }
  }
}

// arity-4, signs (-, +, +, -): 1 thread = 1 clause (one b128 index load).
__global__ void __launch_bounds__(BLK)
clause4_kernel(const float* __restrict__ x, const int* __restrict__ idx,
               float* __restrict__ out, int n) {
  int u = blockIdx.x * BLK + threadIdx.x;
  if (u < n) {
    v4i id = __builtin_nontemporal_load((const v4i*)idx + u);
    float r = fminf(1.0f, 2.0f - x[id.x] + x[id.y] + x[id.z] - x[id.w]);
    __builtin_nontemporal_store(r, out + u);
  }
}

extern "C" void kernel_launch(void* const* d_in, const int* in_sizes, int n_in,
                              void* d_out, int out_size, void* d_ws, size_t ws_size,
                              hipStream_t stream) {
  const float* x    = (const float*)d_in[0];
  const int*   idx0 = (const int*)d_in[1];
  const int*   idx1 = (const int*)d_in[2];
  const int*   idx2 = (const int*)d_in[3];
  float*       out  = (float*)d_out;

  const long L  = in_sizes[0];
  const int  n0 = in_sizes[1] / 2;
  const int  n1 = in_sizes[2] / 3;
  const int  n2 = in_sizes[3] / 4;

  // Warm the 8 MB gather table into L2 ahead of the gather kernels.
  const long tbytes = L * (long)sizeof(float);
  const int  pf_threads = (int)((tbytes + 127) / 128);
  if (pf_threads > 0) {
    prefetch_table_kernel<<<(pf_threads + BLK - 1) / BLK, BLK, 0, stream>>>(x, tbytes);
  }

  const int u0 = (n0 + 1) / 2;
  if (u0 > 0)
    clause2_kernel<<<(u0 + BLK - 1) / BLK, BLK, 0, stream>>>(x, idx0, out, n0);

  const int u1 = (n1 + 3) / 4;
  if (u1 > 0)
    clause3_kernel<<<(u1 + BLK - 1) / BLK, BLK, 0, stream>>>(x, idx1, out + n0, n1);

  if (n2 > 0)
    clause4_kernel<<<(n2 + BLK - 1) / BLK, BLK, 0, stream>>>(x, idx2, out + n0 + n1, n2);
}